// LowRankOrthogonalProjection_48773648613447
// MI455X (gfx1250) — compile-verified
//
#include <hip/hip_runtime.h>

typedef __attribute__((ext_vector_type(2))) float v2f;
typedef __attribute__((ext_vector_type(4))) float v4f;
typedef __attribute__((ext_vector_type(8))) float v8f;

#define EMBED_DIM 4096
#define RANK 64
#define N_TOKENS 8192
#define CHUNK 512
#define NCHUNK (EMBED_DIM / CHUNK)
#define STRIDE_A (CHUNK + 4)   // LDS pad: bank-conflict-free fragment reads
#define STRIDE_T (RANK + 4)

// ---------------------------------------------------------------------------
// Phase 0: WT[r][d] = W[d][r]  (1 MB, one-shot; makes phase-1 B fragments a
// single contiguous global_load_b64 instead of two strided b32 loads)
// ---------------------------------------------------------------------------
__global__ __launch_bounds__(256) void lrproj_transpose(
    const float* __restrict__ W, float* __restrict__ WT)
{
  const int i = blockIdx.x * 256 + threadIdx.x;   // output-linear (coalesced)
  const int r = i >> 12;                          // / EMBED_DIM
  const int d = i & (EMBED_DIM - 1);
  WT[i] = W[(size_t)d * RANK + r];
}

// ---------------------------------------------------------------------------
// Phase 1: T = (source - target) @ W       [N_TOKENS x RANK], fp32 WMMA
// One block (8 waves) per 16-row M tile; double-buffered LDS staging of the
// diff tile; waves split K; ds_add_f32 reduction combines partials.
// ---------------------------------------------------------------------------
__global__ __launch_bounds__(256) void lrproj_phase1(
    const float* __restrict__ src, const float* __restrict__ tgt,
    const float* __restrict__ WT, float* __restrict__ T)
{
  __shared__ float sA[2][16 * STRIDE_A];  // double-buffered diff tile
  __shared__ float sRed[16 * RANK];       // cross-wave reduction buffer

  const int tid  = threadIdx.x;
  const int lane = tid & 31;
  const int wave = tid >> 5;
  const int m0   = blockIdx.x * 16;

  const int row16 = lane & 15;            // M row (A) / N col (B,C) index
  const int koff  = (lane >> 4) << 1;     // 0 (lanes 0-15) or 2 (lanes 16-31)

  v8f acc[4] = {v8f{}, v8f{}, v8f{}, v8f{}};
  v4f dreg[8];

  auto stage_load = [&](int cb) {
    #pragma unroll
    for (int p = 0; p < 8; ++p) {
      const int f  = tid + p * 256;       // float4 index in [0, 16*CHUNK/4)
      const int r  = f >> 7;              // row 0..15
      const int c4 = f & 127;
      const size_t g = (size_t)(m0 + r) * EMBED_DIM + cb + c4 * 4;
      const v4f s = *reinterpret_cast<const v4f*>(&src[g]);
      const v4f t = *reinterpret_cast<const v4f*>(&tgt[g]);
      dreg[p] = s - t;
    }
  };
  auto stage_store = [&](float* buf) {
    #pragma unroll
    for (int p = 0; p < 8; ++p) {
      const int f  = tid + p * 256;
      const int r  = f >> 7;
      const int c4 = f & 127;
      *reinterpret_cast<v4f*>(&buf[r * STRIDE_A + c4 * 4]) = dreg[p];
    }
  };

  stage_load(0);
  stage_store(sA[0]);
  __syncthreads();

  for (int c = 0; c < NCHUNK; ++c) {
    if (c + 1 < NCHUNK) stage_load((c + 1) * CHUNK);   // prefetch next chunk

    const float* buf  = sA[c & 1];
    const int kbase   = wave * 64;       // this wave's K slice of the chunk
    #pragma unroll
    for (int ks = 0; ks < 16; ++ks) {
      const int kk = kbase + ks * 4;
      const v2f a = *reinterpret_cast<const v2f*>(
          &buf[row16 * STRIDE_A + kk + koff]);         // ds_load_b64
      const int kg = c * CHUNK + kk + koff;            // global K of VGPR0
      #pragma unroll
      for (int nt = 0; nt < 4; ++nt) {
        const v2f b = *reinterpret_cast<const v2f*>(
            &WT[(size_t)(nt * 16 + row16) * EMBED_DIM + kg]);  // b64 load
        acc[nt] = __builtin_amdgcn_wmma_f32_16x16x4_f32(
            false, a, false, b, (short)0, acc[nt], false, false);
      }
    }

    if (c + 1 < NCHUNK) stage_store(sA[(c + 1) & 1]);
    __syncthreads();
  }

  // Cross-wave reduction of the 8 partial [16x64] accumulators via ds_add_f32
  for (int i = tid; i < 16 * RANK; i += 256) sRed[i] = 0.0f;
  __syncthreads();
  #pragma unroll
  for (int nt = 0; nt < 4; ++nt) {
    #pragma unroll
    for (int j = 0; j < 8; ++j) {
      const int M = j + ((lane >> 4) << 3);
      const int N = nt * 16 + (lane & 15);
      atomicAdd(&sRed[M * RANK + N], acc[nt][j]);
    }
  }
  __syncthreads();
  for (int i = tid; i < 16 * RANK; i += 256)
    T[(size_t)m0 * RANK + i] = sRed[i];   // coalesced [16x64] store
}

// ---------------------------------------------------------------------------
// Phase 2: out = target + T @ W^T          [N_TOKENS x EMBED_DIM]
// Block = 8 waves covering a 16x128 output tile; T tile staged once in LDS;
// B fragments are contiguous b64 loads straight from rows of W.
// ---------------------------------------------------------------------------
__global__ __launch_bounds__(256) void lrproj_phase2(
    const float* __restrict__ T, const float* __restrict__ W,
    const float* __restrict__ tgt, float* __restrict__ out)
{
  __shared__ float sT[16 * STRIDE_T];

  const int tid  = threadIdx.x;
  const int lane = tid & 31;
  const int wave = tid >> 5;
  const int m0   = blockIdx.y * 16;
  const int n0   = blockIdx.x * 128 + wave * 16;

  // Stage T tile [16 x 64] once per block (shared by all 8 waves)
  #pragma unroll
  for (int p = 0; p < 4; ++p) {
    const int i = tid + p * 256;        // 0..1023
    const int r = i >> 6, c = i & 63;
    sT[r * STRIDE_T + c] = T[(size_t)(m0 + r) * RANK + c];
  }
  __syncthreads();

  const int nl   = lane & 15;
  const int koff = (lane >> 4) << 1;

  v8f acc = {};
  #pragma unroll
  for (int ks = 0; ks < 16; ++ks) {
    const int kk = ks * 4;
    const v2f a = *reinterpret_cast<const v2f*>(
        &sT[nl * STRIDE_T + kk + koff]);               // A = T tile (LDS)
    const v2f b = *reinterpret_cast<const v2f*>(
        &W[(size_t)(n0 + nl) * RANK + kk + koff]);     // B = W^T (row of W)
    acc = __builtin_amdgcn_wmma_f32_16x16x4_f32(
        false, a, false, b, (short)0, acc, false, false);
  }

  // Fused epilogue: out = target + acc
  #pragma unroll
  for (int j = 0; j < 8; ++j) {
    const int M = j + ((lane >> 4) << 3);
    const size_t idx = (size_t)(m0 + M) * EMBED_DIM + n0 + nl;
    out[idx] = tgt[idx] + acc[j];
  }
}

// ---------------------------------------------------------------------------
extern "C" void kernel_launch(void* const* d_in, const int* in_sizes, int n_in,
                              void* d_out, int out_size, void* d_ws, size_t ws_size,
                              hipStream_t stream) {
  (void)in_sizes; (void)n_in; (void)out_size; (void)ws_size;
  const float* src = (const float*)d_in[0];   // [8192 x 4096]
  const float* tgt = (const float*)d_in[1];   // [8192 x 4096]
  const float* W   = (const float*)d_in[2];   // [4096 x 64]
  float* out = (float*)d_out;                 // [8192 x 4096]

  float* T  = (float*)d_ws;                              // [8192 x 64]  (2 MB)
  float* WT = (float*)((char*)d_ws + (size_t)N_TOKENS * RANK * sizeof(float)); // [64 x 4096] (1 MB)

  lrproj_transpose<<<dim3((EMBED_DIM * RANK) / 256), 256, 0, stream>>>(W, WT);
  lrproj_phase1<<<dim3(N_TOKENS / 16), 256, 0, stream>>>(src, tgt, WT, T);
  lrproj_phase2<<<dim3(EMBED_DIM / 128, N_TOKENS / 16), 256, 0, stream>>>(T, W, tgt, out);
}